// ComplementarySystem_52647709114903
// MI455X (gfx1250) — compile-verified
//
#include <hip/hip_runtime.h>
#include <hip/hip_bf16.h>
#include <stdint.h>

// fp32 WMMA vector types for gfx1250 (V_WMMA_F32_16X16X4_F32):
//  A: 16x4 f32  -> 2 VGPRs/lane ; B: 4x16 f32 -> 2 VGPRs/lane ; C/D: 16x16 f32 -> 8 VGPRs/lane
typedef float v2f __attribute__((ext_vector_type(2)));
typedef float v8f __attribute__((ext_vector_type(8)));

// Per-lane element coordinates (wave32):
//  A/B: k_local = elem + 2*(lane>>4), m/n = lane&15
//  C/D: vgpr r -> m = r + 8*(lane>>4), n = lane&15

// gfx1250 async global->LDS copy (ASYNCcnt-tracked). dst VGPR holds the LDS
// byte offset (low 32 bits of the flat shared address), addr is the 64-bit
// global address.
__device__ __forceinline__ void async_g2l_b32(const float* gptr, float* lptr) {
    unsigned           loff = (unsigned)(uintptr_t)lptr;          // LDS byte offset
    unsigned long long ga   = (unsigned long long)(uintptr_t)gptr;
    asm volatile("global_load_async_to_lds_b32 %0, %1, off"
                 :: "v"(loff), "v"(ga) : "memory");
}
__device__ __forceinline__ void wait_async_then_barrier() {
    asm volatile("s_wait_asynccnt 0x0" ::: "memory");
    __syncthreads();
}

// ---------------------------------------------------------------------------
// conv1: 3->64, 7x7, stride 2, SAME (pad_lo=2), + bias + ReLU.
// Implicit GEMM with padded K:  k' = (ci*7+kh)*8 + kw,  kw in 0..7 (kw=7 is a
// zero column in A), K = 21*8 = 168  -> 42 WMMA steps, each chunk within one
// (ci,kh) row.  A staged in LDS [64][168] (zero-padded), B staged as the
// 21 x 40 input-row window (iw_base = 2*ow0-2; cols >=37 zero).
// x chunk: [8][3][224][224], out: [8][64][112][112]
// ---------------------------------------------------------------------------
__global__ void __launch_bounds__(128)
conv1_relu_wmma(const float* __restrict__ x, const float* __restrict__ W,
                const float* __restrict__ bias, float* __restrict__ out) {
    __shared__ float wlds[64 * 168];   // 43.0 KB  (padded weights)
    __shared__ float xlds[21 * 40];    //  3.4 KB  (input rows)

    const int tid  = threadIdx.x;
    const int lane = tid & 31;
    const int wave = tid >> 5;               // 0..3 -> M-tile
    const int ow0  = blockIdx.x * 16;        // 7 * 16 = 112
    const int oh   = blockIdx.y;             // 0..111
    const int img  = blockIdx.z;             // 0..7 (chunk-local)
    const float* __restrict__ ximg = x + (size_t)img * 3 * 224 * 224;

    // ---- stage padded weights: wlds[m][row*8+kw], kw==7 -> 0
    for (int t = tid; t < 64 * 168; t += 128) {
        const int m   = t / 168;
        const int rr  = t - m * 168;
        const int row = rr >> 3;             // ci*7 + kh  (0..20)
        const int kw  = rr & 7;
        if (kw < 7) async_g2l_b32(&W[m * 147 + row * 7 + kw], &wlds[t]);
        else        wlds[t] = 0.0f;
    }
    // ---- stage input rows: xlds[row][c], iw = 2*ow0-2+c, c<37 real, rest 0
    for (int t = tid; t < 21 * 40; t += 128) {
        const int row = t / 40;
        const int c   = t - row * 40;
        const int ci  = row / 7;
        const int kh  = row - ci * 7;
        const int ih  = 2 * oh + kh - 2;     // SAME: pad_lo = 2
        const int iw  = 2 * ow0 - 2 + c;
        const bool ok = (c < 37) & (ih >= 0) & (ih < 224) & (iw >= 0) & (iw < 224);
        if (ok) async_g2l_b32(&ximg[(ci * 224 + ih) * 224 + iw], &xlds[t]);
        else    xlds[t] = 0.0f;
    }
    wait_async_then_barrier();

    const int mA    = wave * 16 + (lane & 15);
    const int n     = lane & 15;
    const int kbase = (lane >> 4) * 2;
    const float* __restrict__ ap  = &wlds[mA * 168 + kbase];   // 8B aligned
    const float* __restrict__ bp0 = &xlds[2 * n + kbase];      // 8B aligned

    v8f acc = {};
    for (int k0 = 0; k0 < 168; k0 += 4) {
        const v2f a = *(const v2f*)(ap + k0);                       // ds_load_b64
        const v2f b = *(const v2f*)(bp0 + (k0 >> 3) * 40 + (k0 & 7)); // ds_load_b64
        acc = __builtin_amdgcn_wmma_f32_16x16x4_f32(
            false, a, false, b, (short)0, acc, false, false);
    }

    const int mbase = wave * 16 + 8 * (lane >> 4);
    const int nout  = ow0 + n;
#pragma unroll
    for (int r = 0; r < 8; ++r) {
        const int m = mbase + r;
        float v = acc[r] + bias[m];
        v = v > 0.0f ? v : 0.0f;
        out[(((size_t)img * 64 + m) * 112 + oh) * 112 + nout] = v;
    }
}

// ---------------------------------------------------------------------------
// conv2: 64->128, 3x3, stride 2, SAME (pad_lo=0, pad_hi=1), + bias + ReLU,
// fused GAP accumulation.  Padded K: k' = (ci*3+kh)*4 + kw, kw in 0..3
// (kw=3 -> A=0), K = 192*4 = 768 -> 192 WMMA steps, one (ci,kh) row each.
// B staged as 192 x 36 input window (iw_base = 2*ow0; cols >=33 zero).
// A read directly from global (row slices are L2-hot).
// h chunk: [8][64][112][112], gap: [64][128]
// ---------------------------------------------------------------------------
__global__ void __launch_bounds__(256)
conv2_gap_wmma(const float* __restrict__ h, const float* __restrict__ W,
               const float* __restrict__ bias, float* __restrict__ gap,
               int img_base) {
    __shared__ float xlds[192 * 36];   // 27.6 KB

    const int tid  = threadIdx.x;
    const int lane = tid & 31;
    const int wave = tid >> 5;               // 0..7 -> M-tile
    const int ow0  = blockIdx.x * 16;        // 4 tiles cover 56 (tail masked)
    const int oh   = blockIdx.y;             // 0..55
    const int img  = blockIdx.z;             // 0..7 (chunk-local)
    const float* __restrict__ himg = h + (size_t)img * 64 * 112 * 112;

    // ---- stage input rows: xlds[row][c], row = ci*3+kh, iw = 2*ow0+c
    for (int t = tid; t < 192 * 36; t += 256) {
        const int row = t / 36;
        const int c   = t - row * 36;
        const int ci  = row / 3;
        const int kh  = row - ci * 3;
        const int ih  = 2 * oh + kh;         // SAME: pad_lo = 0
        const int iw  = 2 * ow0 + c;
        const bool ok = (c < 33) & (ih < 112) & (iw < 112);
        if (ok) async_g2l_b32(&himg[(ci * 112 + ih) * 112 + iw], &xlds[t]);
        else    xlds[t] = 0.0f;
    }
    wait_async_then_barrier();

    const int  mA     = wave * 16 + (lane & 15);
    const int  n      = lane & 15;
    const int  kbase  = (lane >> 4) * 2;
    const bool lohalf = (lane < 16);
    const float* __restrict__ wrow = W + mA * 576;
    const float* __restrict__ bp0  = &xlds[2 * n + kbase];     // 8B aligned

    v8f acc = {};
    for (int r = 0; r < 192; ++r) {          // r = ci*3 + kh
        v2f a;
        a[0] = wrow[r * 3 + kbase];          // kw = 0 (lo) / 2 (hi)
        const float a1 = wrow[r * 3 + 1];    // kw = 1 (valid for lo half)
        a[1] = lohalf ? a1 : 0.0f;           // hi half kw = 3 -> zero pad
        const v2f b = *(const v2f*)(bp0 + r * 36);             // ds_load_b64
        acc = __builtin_amdgcn_wmma_f32_16x16x4_f32(
            false, a, false, b, (short)0, acc, false, false);
    }

    const bool nvalid = (ow0 + n) < 56;
    const int  mbase  = wave * 16 + 8 * (lane >> 4);
    float* __restrict__ grow = gap + (size_t)(img_base + img) * 128;
#pragma unroll
    for (int r = 0; r < 8; ++r) {
        float v = acc[r] + bias[mbase + r];
        v = v > 0.0f ? v : 0.0f;
        v = nvalid ? v : 0.0f;
        // reduce this row's 16 N-columns (lanes share a 16-lane half)
        v += __shfl_xor(v, 1, 32);
        v += __shfl_xor(v, 2, 32);
        v += __shfl_xor(v, 4, 32);
        v += __shfl_xor(v, 8, 32);
        if ((lane & 15) == 0)
            atomicAdd(&grow[mbase + r], v);
    }
}

// ---------------------------------------------------------------------------
// FC (128->2) per branch + argmax decision + emit +/-10 logits.
// gap: [2][64][128] raw sums over 56*56 positions.
// ---------------------------------------------------------------------------
__global__ void fc_decide(const float* __restrict__ gap,
                          const float* __restrict__ tWfc, const float* __restrict__ tbfc,
                          const float* __restrict__ fWfc, const float* __restrict__ fbfc,
                          float* __restrict__ out) {
    const int s = threadIdx.x;
    if (s >= 64) return;
    const float inv = 1.0f / 3136.0f;   // 56*56
    bool spoof = false;
    for (int b = 0; b < 2; ++b) {
        const float* g  = gap + ((size_t)b * 64 + s) * 128;
        const float* Wf = b ? fWfc : tWfc;
        const float* bf = b ? fbfc : tbfc;
        float l0 = bf[0], l1 = bf[1];
        for (int c = 0; c < 128; ++c) {
            const float v = g[c] * inv;
            l0 += v * Wf[c * 2 + 0];
            l1 += v * Wf[c * 2 + 1];
        }
        spoof = spoof || (l1 > l0);     // argmax: ties -> index 0
    }
    out[s * 2 + 0] = spoof ? -10.0f : 10.0f;
    out[s * 2 + 1] = spoof ?  10.0f : -10.0f;
}

__global__ void zero_f32(float* __restrict__ p, int n) {
    const int i = blockIdx.x * blockDim.x + threadIdx.x;
    if (i < n) p[i] = 0.0f;
}

// ---------------------------------------------------------------------------
extern "C" void kernel_launch(void* const* d_in, const int* in_sizes, int n_in,
                              void* d_out, int out_size, void* d_ws, size_t ws_size,
                              hipStream_t stream) {
    const float* x    = (const float*)d_in[0];
    const float* tW1  = (const float*)d_in[1];
    const float* tb1  = (const float*)d_in[2];
    const float* tW2  = (const float*)d_in[3];
    const float* tb2  = (const float*)d_in[4];
    const float* tWfc = (const float*)d_in[5];
    const float* tbfc = (const float*)d_in[6];
    const float* fW1  = (const float*)d_in[7];
    const float* fb1  = (const float*)d_in[8];
    const float* fW2  = (const float*)d_in[9];
    const float* fb2  = (const float*)d_in[10];
    const float* fWfc = (const float*)d_in[11];
    const float* fbfc = (const float*)d_in[12];
    float* out = (float*)d_out;

    // Workspace layout (chunked so the conv1 activation tile stays L2-resident):
    //   buf1: [8][64][112][112] f32 = 25.69 MB
    //   gap : [2][64][128]      f32 = 64 KB
    char*  ws   = (char*)d_ws;
    float* buf1 = (float*)ws;
    const size_t buf1_elems = (size_t)8 * 64 * 112 * 112;
    float* gap  = (float*)(ws + buf1_elems * sizeof(float));

    const int gap_n = 2 * 64 * 128;
    zero_f32<<<(gap_n + 255) / 256, 256, 0, stream>>>(gap, gap_n);

    const float* W1[2] = {tW1, fW1};
    const float* b1[2] = {tb1, fb1};
    const float* W2[2] = {tW2, fW2};
    const float* b2[2] = {tb2, fb2};

    for (int b = 0; b < 2; ++b) {
        for (int ch = 0; ch < 8; ++ch) {   // 8 images per chunk
            const float* xc = x + (size_t)ch * 8 * 3 * 224 * 224;
            conv1_relu_wmma<<<dim3(7, 112, 8), 128, 0, stream>>>(
                xc, W1[b], b1[b], buf1);
            conv2_gap_wmma<<<dim3(4, 56, 8), 256, 0, stream>>>(
                buf1, W2[b], b2[b], gap + (size_t)b * 64 * 128, ch * 8);
        }
    }

    fc_decide<<<1, 64, 0, stream>>>(gap, tWfc, tbfc, fWfc, fbfc, out);
}